// Eq2Net_19679540150893
// MI455X (gfx1250) — compile-verified
//
#include <hip/hip_runtime.h>
#include <math.h>

// Problem constants (match reference)
#define T_LEN 1024
#define NB    64
#define NA    128
#define TP1   1025   // T+1

typedef __attribute__((ext_vector_type(2))) float v2f;
typedef __attribute__((ext_vector_type(8))) float v8f;

// ---- VALU-only wave32 xor-reductions (no DS unit on the critical path) ----
// DPP16 row_xmask: dpp_ctrl = 0x160 | mask  (XOR within each 16-lane row)
template<int CTRL>
__device__ __forceinline__ float dpp_xf(float v) {
  return __int_as_float(
      __builtin_amdgcn_mov_dpp(__float_as_int(v), CTRL, 0xf, 0xf, true));
}
// xor-16 across the two 16-lane halves: v_permlanex16 with identity nibbles
__device__ __forceinline__ float xor16f(float v) {
  int x = __float_as_int(v);
  int r = __builtin_amdgcn_permlanex16(x, x, 0x76543210, 0xfedcba98, false, false);
  return __int_as_float(r);
}
__device__ __forceinline__ float red_max32(float v) {
  v = fmaxf(v, dpp_xf<0x161>(v));   // row_xmask:1
  v = fmaxf(v, dpp_xf<0x162>(v));   // row_xmask:2
  v = fmaxf(v, dpp_xf<0x164>(v));   // row_xmask:4
  v = fmaxf(v, dpp_xf<0x168>(v));   // row_xmask:8
  v = fmaxf(v, xor16f(v));
  return v;
}
__device__ __forceinline__ float red_sum32(float v) {
  v += dpp_xf<0x161>(v);
  v += dpp_xf<0x162>(v);
  v += dpp_xf<0x164>(v);
  v += dpp_xf<0x168>(v);
  v += xor16f(v);
  return v;
}

// ---------------------------------------------------------------------------
// Phase 1: sequential scan, ONE wave32. Lane handles b=lane and b=lane+32.
// Maintains L(b)=logsumexp_{j<=i} u(j,b), C(i,b), running max M(b) of u.
// Emits per step: ws_u[i][b]=u(i,b), ws_lc[i][b]=L_i+C_i,
//                 ws_w[i][b]=M_i+C_i (for phase-2's m_i), ws_c[i][b]=C_i.
// Only ONE 64-wide logsumexp remains on the serial critical path per step.
// ---------------------------------------------------------------------------
__global__ __launch_bounds__(32) void hsmm_scan_kernel(
    const float* __restrict__ stop_logps,   // (T+1, B, 2)
    const float* __restrict__ start_logps,  // (T+1, B)
    float* __restrict__ ws_u, float* __restrict__ ws_lc,
    float* __restrict__ ws_w, float* __restrict__ ws_c)
{
  const int lane = threadIdx.x;
  float L0 = start_logps[lane];
  float L1 = start_logps[lane + 32];
  ws_u[lane] = L0; ws_u[lane + 32] = L1;    // u(0,b) = start_logps[0,b]
  float C0 = 0.f, C1 = 0.f;
  float M0 = L0, M1 = L1;

  for (int i = 1; i < T_LEN; ++i) {
    const float* sp = stop_logps + (size_t)i * NB * 2;
    float stop0 = sp[lane * 2 + 0];
    float cont0 = sp[lane * 2 + 1];
    float stop1 = sp[(lane + 32) * 2 + 0];
    float cont1 = sp[(lane + 32) * 2 + 1];
    float st0 = start_logps[i * NB + lane];
    float st1 = start_logps[i * NB + lane + 32];

    // total_rearrange_i = logsumexp_b(L_{i-1}+C_{i-1}+stop_i) - 0.5
    float t0 = L0 + C0 + stop0;
    float t1 = L1 + C1 + stop1;
    float m  = red_max32(fmaxf(t0, t1));
    float s  = red_sum32(__expf(t0 - m) + __expf(t1 - m));
    float TR = m + __logf(s) - 0.5f;

    C0 += cont0; C1 += cont1;                 // C(i,b)
    float u0 = st0 + TR - C0;                 // u(i,b)
    float u1 = st1 + TR - C1;

    // L = logaddexp(L, u)
    float h0 = fmaxf(L0, u0); L0 = h0 + log1pf(__expf(-fabsf(L0 - u0)));
    float h1 = fmaxf(L1, u1); L1 = h1 + log1pf(__expf(-fabsf(L1 - u1)));
    M0 = fmaxf(M0, u0); M1 = fmaxf(M1, u1);

    ws_u [i * NB + lane]      = u0;
    ws_u [i * NB + lane + 32] = u1;
    ws_lc[i * NB + lane]      = L0 + C0;
    ws_lc[i * NB + lane + 32] = L1 + C1;
    ws_w [i * NB + lane]      = M0 + C0;      // phase 2 computes m_i from this
    ws_w [i * NB + lane + 32] = M1 + C1;
    ws_c [i * NB + lane]      = C0;
    ws_c [i * NB + lane + 32] = C1;
  }
}

// ---------------------------------------------------------------------------
// Phase 2a: all logp terms in parallel. Block i computes one logsumexp over B.
// i=0: logp0; i=1..1023: per-step action terms; i=1024: final stop term.
// ---------------------------------------------------------------------------
__global__ __launch_bounds__(64) void hsmm_logp_kernel(
    const int*   __restrict__ actions,
    const float* __restrict__ action_logps,  // (T+1, B, A)
    const float* __restrict__ stop_logps,    // (T+1, B, 2)
    const float* __restrict__ start_logps,   // (T+1, B)
    const float* __restrict__ ws_lc,
    float* __restrict__ ws_logp)
{
  __shared__ float sh[NB];
  const int i = blockIdx.x;      // 0..1024
  const int b = threadIdx.x;     // 0..63
  float v;
  if (i == 0) {
    v = start_logps[b] + action_logps[(size_t)b * NA + actions[0]];
  } else if (i < T_LEN) {
    v = ws_lc[i * NB + b] + action_logps[((size_t)i * NB + b) * NA + actions[i]];
  } else {
    v = ws_lc[(T_LEN - 1) * NB + b] + stop_logps[((size_t)T_LEN * NB + b) * 2 + 0];
  }
  sh[b] = v; __syncthreads();
  for (int off = 32; off >= 1; off >>= 1) {
    if (b < off) sh[b] = fmaxf(sh[b], sh[b + off]);
    __syncthreads();
  }
  float m = sh[0]; __syncthreads();
  sh[b] = __expf(v - m); __syncthreads();
  for (int off = 32; off >= 1; off >>= 1) {
    if (b < off) sh[b] += sh[b + off];
    __syncthreads();
  }
  if (b == 0) ws_logp[i] = m + __logf(sh[0]);
}

// ---------------------------------------------------------------------------
// Phase 2b (heavy, HBM-bound): block i computes
//   m_i  = max_b ws_w[i][b]           (tight upper bound on dist at step i)
//   cons_i = m_i + log( sum_{j<=i,b} exp( u(j,b) + (C_i(b)-m_i) + cp[j,i,b] ) )
// One pass (all exponents <= 0: cp <= 0 and u+C <= m_i by construction).
// 256 threads: thread = (j-phase, b). Final 256->1 reduction via four
// accumulating V_WMMA_F32_16X16X4_F32 (all-ones A => layout-invariant sum).
// ---------------------------------------------------------------------------
__global__ __launch_bounds__(256) void hsmm_cons_kernel(
    const float* __restrict__ cons_pen,   // (T+1, T+1, B)
    const float* __restrict__ ws_u,
    const float* __restrict__ ws_w,
    const float* __restrict__ ws_c,
    float* __restrict__ ws_cons)
{
  const int i = blockIdx.x + 1;          // 1..1023
  __shared__ float msh[NB];
  __shared__ float zsh[NB];
  __shared__ float partial[256];
  const int t = threadIdx.x;
  if (t < NB) { msh[t] = ws_w[i * NB + t]; zsh[t] = ws_c[i * NB + t]; }
  __syncthreads();
  for (int off = 32; off >= 1; off >>= 1) {
    if (t < off) msh[t] = fmaxf(msh[t], msh[t + off]);
    __syncthreads();
  }
  const float mi = msh[0];
  __syncthreads();
  if (t < NB) zsh[t] -= mi;              // z_i(b) = C_i(b) - m_i
  __syncthreads();

  const int b  = t & 63;
  const int jo = t >> 6;                 // 0..3
  const float zb = zsh[b];
  const float* cp = cons_pen + (size_t)i * NB + b;   // + j*(T+1)*B
  float s = 0.f;
  for (int j = jo; j <= i; j += 4) {
    float x = ws_u[j * NB + b] + zb + cp[(size_t)j * ((size_t)TP1 * NB)];
    s += __expf(x);                       // x <= 0 guaranteed
  }
  partial[t] = s;
  __syncthreads();

  if (t < 32) {                           // wave 0, EXEC all ones
    v2f ones; ones[0] = 1.0f; ones[1] = 1.0f;
    v8f acc = {};
#pragma unroll
    for (int q = 0; q < 4; ++q) {
      v2f bv; bv[0] = partial[q * 64 + 2 * t];
              bv[1] = partial[q * 64 + 2 * t + 1];
      // D = ones(16x4) * B(4x16) + C : column sums; grand total is
      // permutation-invariant under an all-ones A.
      acc = __builtin_amdgcn_wmma_f32_16x16x4_f32(
                false, ones, false, bv, (short)0, acc, false, false);
    }
    float tot = red_sum32(acc[0]);        // = 2 * grand total
    if (t == 0) ws_cons[i] = mi + __logf(tot * 0.5f);
  }
}

// ---------------------------------------------------------------------------
// Phase 3: deterministic serial combine of ~2K scalars.
// ---------------------------------------------------------------------------
__global__ void hsmm_final_kernel(const float* __restrict__ ws_logp,
                                  const float* __restrict__ ws_cons,
                                  float* __restrict__ out)
{
  if (threadIdx.x == 0 && blockIdx.x == 0) {
    float logp = 0.f;
    for (int i = 0; i <= T_LEN; ++i) logp += ws_logp[i];
    float cons = 0.f;
    for (int i = 1; i < T_LEN; ++i) cons += ws_cons[i];
    out[0] = -logp + cons;
  }
}

extern "C" void kernel_launch(void* const* d_in, const int* in_sizes, int n_in,
                              void* d_out, int out_size, void* d_ws, size_t ws_size,
                              hipStream_t stream) {
  const int*   actions      = (const int*)  d_in[0];
  const float* action_logps = (const float*)d_in[1];
  const float* stop_logps   = (const float*)d_in[2];
  const float* start_logps  = (const float*)d_in[3];
  const float* cons_pen     = (const float*)d_in[4];
  float* out = (float*)d_out;

  float* ws      = (float*)d_ws;
  float* ws_u    = ws;                //  65536 floats
  float* ws_lc   = ws + 65536;        //  65536
  float* ws_w    = ws + 131072;       //  65536
  float* ws_c    = ws + 196608;       //  65536
  float* ws_logp = ws + 262144;       //   1025
  float* ws_cons = ws + 263296;       //   1024   (total ~1.03 MB)

  hipLaunchKernelGGL(hsmm_scan_kernel, dim3(1), dim3(32), 0, stream,
                     stop_logps, start_logps, ws_u, ws_lc, ws_w, ws_c);
  hipLaunchKernelGGL(hsmm_logp_kernel, dim3(TP1), dim3(64), 0, stream,
                     actions, action_logps, stop_logps, start_logps, ws_lc, ws_logp);
  hipLaunchKernelGGL(hsmm_cons_kernel, dim3(T_LEN - 1), dim3(256), 0, stream,
                     cons_pen, ws_u, ws_w, ws_c, ws_cons);
  hipLaunchKernelGGL(hsmm_final_kernel, dim3(1), dim3(1), 0, stream,
                     ws_logp, ws_cons, out);
}